// GCNModel_37031208026375
// MI455X (gfx1250) — compile-verified
//
#include <hip/hip_runtime.h>

typedef __attribute__((ext_vector_type(2))) float v2f;
typedef __attribute__((ext_vector_type(8))) float v8f;

#define F 128  // feature width of the GNN layers

// ---------------------------------------------------------------- utilities
__global__ void fill_zero_kernel(float* p, int n) {
  int i = blockIdx.x * blockDim.x + threadIdx.x;
  if (i < n) p[i] = 0.0f;
}

__global__ void deg_kernel(const int* __restrict__ col, float* __restrict__ deg, int E) {
  int e = blockIdx.x * blockDim.x + threadIdx.x;
  if (e < E)
    __hip_atomic_fetch_add(&deg[col[e]], 1.0f, __ATOMIC_RELAXED, __HIP_MEMORY_SCOPE_AGENT);
}

__global__ void dinv_kernel(const float* __restrict__ deg, float* __restrict__ dinv, int N) {
  int i = blockIdx.x * blockDim.x + threadIdx.x;
  if (i < N) dinv[i] = rsqrtf(deg[i] + 2.0f);  // deg+2 > 0 always
}

// ------------------------------------------------ WMMA GEMM: out = X @ W
// X: [nrows,128], W: [128,128] row-major, out: [nrows,128]. fp32 WMMA 16x16x4.
// Block: 256 threads = 8 waves; each wave computes a 16x128 row slab.
__global__ __launch_bounds__(256) void gemm128_wmma_kernel(
    const float* __restrict__ X, const float* __restrict__ W,
    float* __restrict__ out, int nrows) {
  // W staged as k-pair float2: WP[(k>>1)*128 + n] = {W[k][n], W[k+1][n]}  (64KB)
  __shared__ float2 WP[64 * F];
  for (int i = threadIdx.x; i < 64 * F; i += 256) {
    int kh = i >> 7;        // k/2
    int n  = i & 127;
    float2 w;
    w.x = W[(2 * kh) * F + n];
    w.y = W[(2 * kh + 1) * F + n];
    WP[i] = w;
  }
  __syncthreads();

  const int lane  = threadIdx.x & 31;
  const int wave  = threadIdx.x >> 5;
  const int mbase = (blockIdx.x * 8 + wave) * 16;
  const int half  = lane >> 4;        // 0: K=k0,k0+1  1: K=k0+2,k0+3
  const int colq  = lane & 15;

  int arow = mbase + colq;            // A-frag row for this lane (M = lane%16)
  if (arow >= nrows) arow = nrows - 1;  // clamp (keeps EXEC full for WMMA)
  const float* Xr = X + (size_t)arow * F;

  v8f acc[8] = {};  // 16x128 fp32 accumulator tile

  for (int k0 = 0; k0 < F; k0 += 4) {
    // A fragment: 16x4 f32 -> 2 VGPRs/lane (one b64 load)
    float2 av = *(const float2*)(Xr + k0 + (half << 1));
    v2f a; a.x = av.x; a.y = av.y;
    const int kp = (k0 >> 1) + half;  // k-pair index in LDS
#pragma unroll
    for (int n = 0; n < 8; ++n) {
      float2 bv = WP[kp * F + n * 16 + colq];   // 4x16 f32 B fragment
      v2f b; b.x = bv.x; b.y = bv.y;
      acc[n] = __builtin_amdgcn_wmma_f32_16x16x4_f32(
          false, a, false, b, (short)0, acc[n], false, false);
    }
  }

  // C/D layout: VGPR v -> M = mbase + v + 8*(lane>=16), N = n*16 + lane%16
  const int rofs = mbase + (half << 3);
#pragma unroll
  for (int n = 0; n < 8; ++n) {
#pragma unroll
    for (int v = 0; v < 8; ++v) {
      int r = rofs + v;
      if (r < nrows) out[(size_t)r * F + n * 16 + colq] = acc[n][v];
    }
  }
}

// -------------------------------------- agg = b + 2*dinv^2 * xw  (self loop)
__global__ void init_agg_kernel(const float* __restrict__ xw,
                                const float* __restrict__ dinv,
                                const float* __restrict__ b,
                                float* __restrict__ agg, long n4) {
  long t = (long)blockIdx.x * blockDim.x + threadIdx.x;  // over N*32 float4s
  if (t >= n4) return;
  int node = (int)(t >> 5);
  int q    = (int)(t & 31);
  float s = dinv[node];
  s = 2.0f * s * s;
  float4 v  = ((const float4*)xw)[t];
  float4 bb = ((const float4*)b)[q];
  float4 o;
  o.x = bb.x + s * v.x; o.y = bb.y + s * v.y;
  o.z = bb.z + s * v.z; o.w = bb.w + s * v.w;
  ((float4*)agg)[t] = o;
}

// ---------------- edge scatter: agg[col] += dinv[row]*dinv[col]*xw[row]
// One wave per edge; 32 lanes x float4 = full 128-float feature row.
__global__ __launch_bounds__(256) void scatter_kernel(
    const int* __restrict__ row, const int* __restrict__ col,
    const float* __restrict__ dinv, const float* __restrict__ xw,
    float* __restrict__ agg, int E) {
  int wid  = threadIdx.x >> 5;
  int lane = threadIdx.x & 31;
  int e = blockIdx.x * 8 + wid;
  if (e >= E) return;
  int r = row[e];
  int c = col[e];
  float norm = dinv[r] * dinv[c];
  float4 v = ((const float4*)(xw + (size_t)r * F))[lane];   // L2-resident gather
  float* dst = agg + (size_t)c * F + lane * 4;
  __hip_atomic_fetch_add(dst + 0, norm * v.x, __ATOMIC_RELAXED, __HIP_MEMORY_SCOPE_AGENT);
  __hip_atomic_fetch_add(dst + 1, norm * v.y, __ATOMIC_RELAXED, __HIP_MEMORY_SCOPE_AGENT);
  __hip_atomic_fetch_add(dst + 2, norm * v.z, __ATOMIC_RELAXED, __HIP_MEMORY_SCOPE_AGENT);
  __hip_atomic_fetch_add(dst + 3, norm * v.w, __ATOMIC_RELAXED, __HIP_MEMORY_SCOPE_AGENT);
}

__global__ void relu4_kernel(float4* p, long n4) {
  long t = (long)blockIdx.x * blockDim.x + threadIdx.x;
  if (t >= n4) return;
  float4 v = p[t];
  v.x = fmaxf(v.x, 0.0f); v.y = fmaxf(v.y, 0.0f);
  v.z = fmaxf(v.z, 0.0f); v.w = fmaxf(v.w, 0.0f);
  p[t] = v;
}

// -------- relu + dropout(*mask*2) + segment_max -> pooled[64,128]
// all values >= 0, so integer-bit atomicMax == float max; pooled pre-zeroed.
__global__ void pool_kernel(const float* __restrict__ agg,
                            const int* __restrict__ mask,
                            const int* __restrict__ batch,
                            float* __restrict__ pooled, long total) {
  long t = (long)blockIdx.x * blockDim.x + threadIdx.x;  // over N*128
  if (t >= total) return;
  int node = (int)(t >> 7);
  float v = fmaxf(agg[t], 0.0f) * (float)mask[t] * 2.0f;
  int g = batch[node];
  atomicMax((int*)&pooled[(size_t)g * F + (t & 127)], __float_as_int(v));
}

// -------- tail MLP (tiny: 64 rows)
__global__ void mlp1_kernel(const float* __restrict__ pooled,
                            const float* __restrict__ Wm,
                            const float* __restrict__ bm,
                            float* __restrict__ pm) {
  int t = blockIdx.x * blockDim.x + threadIdx.x;  // 64*128
  if (t >= 64 * F) return;
  int g = t >> 7, j = t & 127;
  float s = bm[j];
  const float* pr = pooled + (size_t)g * F;
  for (int k = 0; k < F; ++k) s += pr[k] * Wm[k * F + j];
  pm[t] = fmaxf(s, 0.0f);
}

__global__ void mlp2_kernel(const float* __restrict__ pm,
                            const float* __restrict__ Wf,
                            const float* __restrict__ bf,
                            float* __restrict__ out) {
  int t = blockIdx.x * blockDim.x + threadIdx.x;  // 64*32
  if (t >= 64 * 32) return;
  int g = t >> 5, j = t & 31;
  float s = bf[j];
  const float* pr = pm + (size_t)g * F;
  for (int k = 0; k < F; ++k) s += pr[k] * Wf[k * 32 + j];
  out[t] = s;
}

// ---------------------------------------------------------------- launcher
extern "C" void kernel_launch(void* const* d_in, const int* in_sizes, int n_in,
                              void* d_out, int out_size, void* d_ws, size_t ws_size,
                              hipStream_t stream) {
  const float* x     = (const float*)d_in[0];
  const int*   edge  = (const int*)d_in[1];
  const int*   batch = (const int*)d_in[2];
  const int*   mask  = (const int*)d_in[3];
  const float* W1 = (const float*)d_in[4];
  const float* b1 = (const float*)d_in[5];
  const float* W2 = (const float*)d_in[6];
  const float* b2 = (const float*)d_in[7];
  const float* Wm = (const float*)d_in[8];
  const float* bm = (const float*)d_in[9];
  const float* Wf = (const float*)d_in[10];
  const float* bf = (const float*)d_in[11];
  float* out = (float*)d_out;

  const int N = in_sizes[0] / F;       // 100000 nodes
  const int E = in_sizes[1] / 2;       // 1600000 edges
  const int* rowi = edge;
  const int* coli = edge + E;

  // workspace layout (floats)
  float* ws     = (float*)d_ws;
  float* deg    = ws;                               // N
  float* dinv   = deg + N;                          // N
  float* bufA   = dinv + N;                         // N*128  (xw)
  float* bufB   = bufA + (size_t)N * F;             // N*128  (agg / h)
  float* pooled = bufB + (size_t)N * F;             // 64*128
  float* pm     = pooled + 64 * F;                  // 64*128

  const long n4    = (long)N * (F / 4);             // float4 count per node buf
  const long total = (long)N * F;
  const int  gblocks = (N + 127) / 128;

  // degrees + normalization
  fill_zero_kernel<<<(N + 255) / 256, 256, 0, stream>>>(deg, N);
  deg_kernel<<<(E + 255) / 256, 256, 0, stream>>>(coli, deg, E);
  dinv_kernel<<<(N + 255) / 256, 256, 0, stream>>>(deg, dinv, N);

  // ---- layer 1
  gemm128_wmma_kernel<<<gblocks, 256, 0, stream>>>(x, W1, bufA, N);
  init_agg_kernel<<<(int)((n4 + 255) / 256), 256, 0, stream>>>(bufA, dinv, b1, bufB, n4);
  scatter_kernel<<<(E + 7) / 8, 256, 0, stream>>>(rowi, coli, dinv, bufA, bufB, E);
  relu4_kernel<<<(int)((n4 + 255) / 256), 256, 0, stream>>>((float4*)bufB, n4);

  // ---- layer 2
  gemm128_wmma_kernel<<<gblocks, 256, 0, stream>>>(bufB, W2, bufA, N);
  init_agg_kernel<<<(int)((n4 + 255) / 256), 256, 0, stream>>>(bufA, dinv, b2, bufB, n4);
  scatter_kernel<<<(E + 7) / 8, 256, 0, stream>>>(rowi, coli, dinv, bufA, bufB, E);

  // ---- dropout + global max pool
  fill_zero_kernel<<<(64 * F + 255) / 256, 256, 0, stream>>>(pooled, 64 * F);
  pool_kernel<<<(int)((total + 255) / 256), 256, 0, stream>>>(bufB, mask, batch, pooled, total);

  // ---- tail MLP
  mlp1_kernel<<<(64 * F + 255) / 256, 256, 0, stream>>>(pooled, Wm, bm, pm);
  mlp2_kernel<<<(64 * 32 + 255) / 256, 256, 0, stream>>>(pm, Wf, bf, out);
}